// MusicalGrammarLoss_86818468921556
// MI455X (gfx1250) — compile-verified
//
#include <hip/hip_runtime.h>
#include <cstdint>

typedef float v4f __attribute__((ext_vector_type(4)));

constexpr int B_SEQ = 16;
constexpr int S_LEN = 2048;
constexpr int VOCAB = 804;
constexpr int NV4   = VOCAB / 4;          // 201 float4 per row (16B-aligned rows)
constexpr int ROWS  = B_SEQ * S_LEN;      // 32768
constexpr int WPB   = 8;                  // waves per block in argmax kernel

// ---------------------------------------------------------------------------
// Kernel 1: row-wise argmax over 804 fp32, first-index tie-break.
// Bandwidth-bound: 105.6MB streamed with non-temporal b128 loads.
// ---------------------------------------------------------------------------
__global__ __launch_bounds__(256) void k_argmax(const float* __restrict__ pred,
                                                int* __restrict__ ids) {
  const int wave = threadIdx.x >> 5;
  const int lane = threadIdx.x & 31;
  const int row  = blockIdx.x * WPB + wave;
  if (row >= ROWS) return;

  const v4f* rp = (const v4f*)(pred + (size_t)row * VOCAB);

  float bestV = -__builtin_inff();
  int   bestI = 0;

  v4f v[6];
#pragma unroll
  for (int k = 0; k < 6; ++k)
    v[k] = __builtin_nontemporal_load(rp + lane + 32 * k);   // th:NT streaming
  const bool tail = lane < (NV4 - 192);                      // lanes 0..8
  v4f vt = {};
  if (tail) vt = __builtin_nontemporal_load(rp + lane + 192);

#pragma unroll
  for (int k = 0; k < 6; ++k) {
    const int base = (lane + 32 * k) * 4;                    // ascending per lane
    if (v[k].x > bestV) { bestV = v[k].x; bestI = base + 0; }
    if (v[k].y > bestV) { bestV = v[k].y; bestI = base + 1; }
    if (v[k].z > bestV) { bestV = v[k].z; bestI = base + 2; }
    if (v[k].w > bestV) { bestV = v[k].w; bestI = base + 3; }
  }
  if (tail) {
    const int base = (lane + 192) * 4;
    if (vt.x > bestV) { bestV = vt.x; bestI = base + 0; }
    if (vt.y > bestV) { bestV = vt.y; bestI = base + 1; }
    if (vt.z > bestV) { bestV = vt.z; bestI = base + 2; }
    if (vt.w > bestV) { bestV = vt.w; bestI = base + 3; }
  }

  // wave32 butterfly reduction, tie -> lower index (matches jnp.argmax)
#pragma unroll
  for (int off = 16; off > 0; off >>= 1) {
    const float ov = __shfl_xor(bestV, off, 32);
    const int   oi = __shfl_xor(bestI, off, 32);
    if (ov > bestV || (ov == bestV && oi < bestI)) { bestV = ov; bestI = oi; }
  }
  if (lane == 0) ids[row] = bestI;
}

// ---------------------------------------------------------------------------
// Pairing automaton step (per-pitch, all state in registers)
// ---------------------------------------------------------------------------
__device__ __forceinline__ void pair_step(int tok, int myPitch, int& cnt, int& last,
                                          float& orph, float& adjc) {
  const bool is_on  = (unsigned)(tok - 4)   < 128u;   // NOTE_ON
  const bool is_off = (unsigned)(tok - 132) < 128u;   // NOTE_OFF
  if (is_on | is_off) {
    const int p = tok - (is_on ? 4 : 132);
    if (p == myPitch) {
      if (is_on) {
        if (last == 1) adjc += 1.0f;        // on after on
        ++cnt; last = 1;
      } else if (cnt > 0) {
        if (last == 2) adjc += 1.0f;        // paired-off after paired-off
        --cnt; last = 2;
      } else {
        orph += 1.0f; last = 3;             // orphan NOTE_OFF
      }
    }
  }
}

// ---------------------------------------------------------------------------
// Kernel 2: one block per sequence (0..15 pred, 16..31 target), 128 threads.
// Tokens staged into LDS via gfx1250 async global->LDS (ASYNCcnt path).
// Thread p runs the pairing automaton for pitch p entirely in registers.
// ---------------------------------------------------------------------------
__global__ __launch_bounds__(128) void k_grammar(const int* __restrict__ pred_ids,
                                                 const int* __restrict__ tgt_ids,
                                                 float* __restrict__ partials) {
  __shared__ int   toks[S_LEN];
  __shared__ float fPair[128];
  __shared__ float fVel[128];
  __shared__ float fTs[128];
  __shared__ int   iFlags[128];   // bit0 hasNote, bit1 hasTs
  __shared__ int   sFirst[128], sLastT[128], sPref[128], sSuff[128], sBest[128], sUni[128];

  const int  b      = blockIdx.x;          // 0..31
  const int  t      = threadIdx.x;         // 0..127
  const bool isPred = (b < B_SEQ);
  const int* src    = isPred ? (pred_ids + b * S_LEN) : (tgt_ids + (b - B_SEQ) * S_LEN);

  // ---- async stage: 2048 ints -> LDS, fully coalesced b128 per lane ----
#pragma unroll
  for (int k = 0; k < 4; ++k) {
    const int idx = (k * 128 + t) * 4;                       // 4 ints per lane
    const unsigned la = (unsigned)(uintptr_t)(&toks[idx]);   // LDS byte offset
    const unsigned long long ga = (unsigned long long)(uintptr_t)(src + idx);
    asm volatile("global_load_async_to_lds_b128 %0, %1, off"
                 :: "v"(la), "v"(ga) : "memory");
  }
  asm volatile("s_wait_asynccnt 0" ::: "memory");            // ASYNCcnt drain
  __syncthreads();

  // ---- per-pitch pairing automaton (thread t == pitch t), registers only ----
  int   cnt = 0, last = 0;
  float orph = 0.f, adjc = 0.f;
  for (int i = 0; i < S_LEN; i += 4) {
    const int t0 = toks[i], t1 = toks[i + 1], t2 = toks[i + 2], t3 = toks[i + 3];
    pair_step(t0, t, cnt, last, orph, adjc);
    pair_step(t1, t, cnt, last, orph, adjc);
    pair_step(t2, t, cnt, last, orph, adjc);
    pair_step(t3, t, cnt, last, orph, adjc);
  }
  fPair[t] = 2.0f * orph + 0.5f * adjc + 1.5f * (float)cnt;

  // ---- elementwise penalties + run summaries over this thread's 16 tokens ----
  const int c0 = t * 16;
  float velS = 0.f, tsS = 0.f;
  int flags = 0;
  {
    int prev = toks[c0];
    int run = 1, best = 1, pref = 1;
    bool prefAlive = true;
    // first token classification
    {
      const int tok = prev;
      if (tok >= 4 && tok < 260) flags |= 1;
      if (tok >= 292 && tok < VOCAB) {
        flags |= 2;
        const float dur = (float)(tok - 292) * 15.625f / 1000.0f;
        tsS += (dur <= 0.01f) ? 2.0f : (dur <= 0.05f) ? 0.5f : (dur <= 0.1f) ? 0.1f : 0.0f;
      }
      if (tok >= 260 && tok < 292) {
        const int midi = (int)floorf((float)(tok - 260) / 31.0f * 126.0f) + 1;
        velS += (midi <= 20) ? 3.0f : (midi <= 40) ? 1.0f : (midi <= 63) ? 0.2f : 0.0f;
      }
    }
    sFirst[t] = prev;
    for (int j = 1; j < 16; ++j) {
      const int tok = toks[c0 + j];
      if (tok == prev) { ++run; if (run > best) best = run; }
      else             { run = 1; prefAlive = false; }
      if (prefAlive) ++pref;
      prev = tok;
      if (tok >= 4 && tok < 260) flags |= 1;
      if (tok >= 292 && tok < VOCAB) {
        flags |= 2;
        const float dur = (float)(tok - 292) * 15.625f / 1000.0f;
        tsS += (dur <= 0.01f) ? 2.0f : (dur <= 0.05f) ? 0.5f : (dur <= 0.1f) ? 0.1f : 0.0f;
      }
      if (tok >= 260 && tok < 292) {
        const int midi = (int)floorf((float)(tok - 260) / 31.0f * 126.0f) + 1;
        velS += (midi <= 20) ? 3.0f : (midi <= 40) ? 1.0f : (midi <= 63) ? 0.2f : 0.0f;
      }
    }
    sLastT[t] = prev; sPref[t] = pref; sSuff[t] = run; sBest[t] = best;
    sUni[t] = prefAlive ? 1 : 0;
  }
  fVel[t] = velS; fTs[t] = tsS; iFlags[t] = flags;
  __syncthreads();

  // ---- thread 0: deterministic block reductions -> 5 partials ----
  if (t == 0) {
    float pairSum = 0.f, velSum = 0.f, tsSum = 0.f;
    int allFlags = 0;
    for (int i = 0; i < 128; ++i) {
      pairSum += fPair[i]; velSum += fVel[i]; tsSum += fTs[i]; allFlags |= iFlags[i];
    }
    // merge run summaries left-to-right
    int accBest = sBest[0], accSuff = sSuff[0], accLast = sLastT[0];
    for (int i = 1; i < 128; ++i) {
      const bool joined = (accLast == sFirst[i]);
      if (joined) {
        const int cross = accSuff + sPref[i];
        if (cross > accBest) accBest = cross;
        accSuff = sUni[i] ? (accSuff + 16) : sSuff[i];
      } else {
        accSuff = sSuff[i];
      }
      if (sBest[i] > accBest) accBest = sBest[i];
      accLast = sLastT[i];
    }
    const int maxRep = accBest;                               // 1 + max streak
    const float repPen = (maxRep > 3) ? (float)((maxRep - 3) * (maxRep - 3)) : 0.f;
    const float coh = ((allFlags & 1) ? 0.f : 5.f) + ((allFlags & 2) ? 0.f : 2.f);

    partials[b * 5 + 0] = pairSum;                            // pairing (pred + tgt)
    partials[b * 5 + 1] = isPred ? coh    : 0.f;
    partials[b * 5 + 2] = isPred ? repPen : 0.f;
    partials[b * 5 + 3] = isPred ? velSum : 0.f;
    partials[b * 5 + 4] = isPred ? tsSum  : 0.f;
  }
}

// ---------------------------------------------------------------------------
// Kernel 3: final scalar combine (reference weighting & division order)
// ---------------------------------------------------------------------------
__global__ void k_final(const float* __restrict__ partials,
                        const float* __restrict__ base_loss,
                        float* __restrict__ out) {
  float s0 = 0.f, s1 = 0.f, s2 = 0.f, s3 = 0.f, s4 = 0.f;
  for (int b = 0; b < 2 * B_SEQ; ++b) {
    s0 += partials[b * 5 + 0];
    s1 += partials[b * 5 + 1];
    s2 += partials[b * 5 + 2];
    s3 += partials[b * 5 + 3];
    s4 += partials[b * 5 + 4];
  }
  const float Bf = (float)B_SEQ;
  out[0] = base_loss[0]
         + 10.0f * (s0 / Bf)   // note pairing
         +  5.0f * (s1 / Bf)   // coherence
         +  3.0f * (s2 / Bf)   // repetition
         +  5.0f * (s3 / Bf)   // velocity
         +  3.0f * (s4 / Bf);  // timing
}

// ---------------------------------------------------------------------------
extern "C" void kernel_launch(void* const* d_in, const int* in_sizes, int n_in,
                              void* d_out, int out_size, void* d_ws, size_t ws_size,
                              hipStream_t stream) {
  const float* pred = (const float*)d_in[0];   // [16,2048,804] fp32
  const int*   tgt  = (const int*)d_in[1];     // [16,2048] int32
  const float* base = (const float*)d_in[2];   // scalar fp32

  int*   ids      = (int*)d_ws;                                    // 32768 ints
  float* partials = (float*)((char*)d_ws + (size_t)ROWS * sizeof(int)); // 32*5 floats

  k_argmax <<<ROWS / WPB, 256, 0, stream>>>(pred, ids);
  k_grammar<<<2 * B_SEQ, 128, 0, stream>>>(ids, tgt, partials);
  k_final  <<<1, 1, 0, stream>>>(partials, base, (float*)d_out);
}